// AttentionBasedAggregation_13838384628101
// MI455X (gfx1250) — compile-verified
//
#include <hip/hip_runtime.h>

typedef float v2f __attribute__((ext_vector_type(2)));
typedef float v8f __attribute__((ext_vector_type(8)));

constexpr int kTotal   = 262144;
constexpr int kD       = 128;   // feature dim
constexpr int kH       = 8;     // heads
constexpr int kB       = 64;    // segments
constexpr int kSplits  = 32;    // blocks per segment
constexpr int kTiles   = kD / 16;  // 8 N-tiles of 16 dims
constexpr int kWaves   = 8;
constexpr int kThreads = kWaves * 32;

// ---------------------------------------------------------------------------
// Zero the f32 partial-sum workspace (graph-capture-safe; no hipMemsetAsync).
// ---------------------------------------------------------------------------
__global__ void zero_ws_kernel(float* __restrict__ ws, int n) {
  int i = blockIdx.x * blockDim.x + threadIdx.x;
  if (i < n) ws[i] = 0.0f;
}

// ---------------------------------------------------------------------------
// Main kernel: per-segment attᵀ(n×8) × features(n×128) via f32 WMMA 16x16x4.
// One block = (segment, split). Rows are sorted by segment id, so each block
// covers rows of exactly one segment -> WMMA accumulators never mix segments.
// Hot loop is fully branchless (clamped addresses + mask-multiply zeroing of
// the A operand) and stages ALL 18 loads of a K-step into distinct registers
// before the WMMA chain, so the scheduler can issue one load clause and
// stagger s_wait_loadcnt across the 8 WMMAs (max memory-level parallelism).
// ---------------------------------------------------------------------------
__global__ __launch_bounds__(kThreads) void seg_wmma_kernel(
    const float* __restrict__ feat,   // [kTotal, kD]
    const float* __restrict__ att,    // [kTotal, kH]
    const int*   __restrict__ seg,    // [kTotal], sorted
    float*       __restrict__ ws) {   // [kB*kH*kD] partials + [kB*kH] wsums
  __shared__ float lds_acc[kH * kD];  // 4 KB block accumulator (head, dim)
  __shared__ float lds_ws[kH];
  __shared__ int   bounds[2];

  const int tid  = threadIdx.x;
  const int lane = tid & 31;
  const int wave = tid >> 5;
  const int b    = blockIdx.x >> 5;            // segment id
  const int s    = blockIdx.x & (kSplits - 1); // split within segment

  for (int i = tid; i < kH * kD; i += kThreads) lds_acc[i] = 0.0f;
  if (tid < kH) lds_ws[tid] = 0.0f;

  if (tid == 0) {
    // lower_bound(b) and lower_bound(b+1) on sorted segment ids
    int lo = 0, hi = kTotal;
    while (lo < hi) { int mid = (lo + hi) >> 1; if (seg[mid] <  b) lo = mid + 1; else hi = mid; }
    bounds[0] = lo;
    hi = kTotal;
    while (lo < hi) { int mid = (lo + hi) >> 1; if (seg[mid] <= b) lo = mid + 1; else hi = mid; }
    bounds[1] = lo;
  }
  __syncthreads();

  const int segStart = bounds[0];
  const int segEnd   = bounds[1];
  const int len      = segEnd - segStart;
  const int chunk    = (len + kSplits - 1) / kSplits;
  const int cs       = segStart + s * chunk;
  const int ce       = min(cs + chunk, segEnd);   // uniform across block

  if (cs < ce) {
    v8f acc[kTiles] = {};      // 8 accumulators x 8 VGPRs = 16(M) x 128(N) f32
    float ws_acc = 0.0f;       // per-lane att sum (head = lane&15)

    const int   halfK  = lane >> 4;   // 0: K rows 0,1   1: K rows 2,3
    const int   m      = lane & 15;   // A: head index / B: dim-within-tile
    const int   hclamp = m & (kH - 1);
    const float headok = (m < kH) ? 1.0f : 0.0f;   // heads 8..15 zero-padded

    const int nsteps = (ce - cs + 3) >> 2;   // K-steps of 4 rows
    for (int i = wave; i < nsteps; i += kWaves) {
      const int k0 = cs + (i << 2);
      const int r0 = k0 + 2 * halfK;        // this lane's row for A.x / B.x
      const int r1 = r0 + 1;                // this lane's row for A.y / B.y
      const int r0c = min(r0, kTotal - 1);  // clamped rows are always valid,
      const int r1c = min(r1, kTotal - 1);  // finite data (no 0*Inf -> NaN)

      // ---- Load phase: issue all 18 loads into distinct registers --------
      const float av0 = __builtin_nontemporal_load(&att[(size_t)r0c * kH + hclamp]);
      const float av1 = __builtin_nontemporal_load(&att[(size_t)r1c * kH + hclamp]);

      const float* f0 = feat + (size_t)r0c * kD + m;
      const float* f1 = feat + (size_t)r1c * kD + m;
      v2f Bv[kTiles];
#pragma unroll
      for (int t = 0; t < kTiles; ++t) {
        Bv[t].x = __builtin_nontemporal_load(f0 + t * 16);
        Bv[t].y = __builtin_nontemporal_load(f1 + t * 16);
      }

      // Branchless prefetch of this wave's next K-step row (clamped, valid).
      const int pk = min(k0 + (kWaves << 2), kTotal - 1);
      __builtin_prefetch(&feat[(size_t)pk * kD], 0, 3);

      // ---- A-operand masking (overlaps with loads in flight) -------------
      // Zero via mask MULTIPLY so the att loads are unconditionally used and
      // cannot be sunk into divergent branches (EXEC stays all-1s for WMMA).
      const float m0 = (r0 < ce) ? headok : 0.0f;
      const float m1 = (r1 < ce) ? headok : 0.0f;
      const float a0 = av0 * m0;
      const float a1 = av1 * m1;
      ws_acc += a0 + a1;
      v2f A; A.x = a0; A.y = a1;

      // ---- WMMA phase: D = A(16x4) * B(4x16) + C (f32, K=4) --------------
#pragma unroll
      for (int t = 0; t < kTiles; ++t) {
        acc[t] = __builtin_amdgcn_wmma_f32_16x16x4_f32(
            /*neg_a=*/false, A, /*neg_b=*/false, Bv[t],
            /*c_mod=*/(short)0, acc[t], /*reuse_a=*/false, /*reuse_b=*/false);
      }
    }

    // Wave -> LDS. C/D layout: VGPR v, lanes 0-15 hold M=v (heads 0..7, all
    // valid), lanes 16-31 hold M=v+8 (padding heads) -> skip.
    if (lane < 16) {
#pragma unroll
      for (int t = 0; t < kTiles; ++t) {
#pragma unroll
        for (int v = 0; v < 8; ++v) {
          atomicAdd(&lds_acc[v * kD + t * 16 + lane], acc[t][v]);  // ds_add_f32
        }
      }
    }
    if (m < kH) atomicAdd(&lds_ws[m], ws_acc);  // both half-waves contribute
  }
  __syncthreads();

  // Block -> global partials (global_atomic_add_f32, ~8 MB total traffic).
  if (cs < ce) {
    for (int i = tid; i < kH * kD; i += kThreads)
      atomicAdd(&ws[b * (kH * kD) + i], lds_acc[i]);
    if (tid < kH)
      atomicAdd(&ws[kB * kH * kD + b * kH + tid], lds_ws[tid]);
  }
}

// ---------------------------------------------------------------------------
// Finalize: avg = (denom==0) ? 0 : num/denom;  NaN -> 1e-5. Also emit wsums.
// d_out layout: [kB*kH*kD] avg, then [kB*kH] weights_sum (flat, return order).
// ---------------------------------------------------------------------------
__global__ void finalize_kernel(const float* __restrict__ ws,
                                float* __restrict__ out) {
  const int nAvg = kB * kH * kD;
  int i = blockIdx.x * blockDim.x + threadIdx.x;
  if (i < nAvg) {
    const int b = i / (kH * kD);
    const int h = (i / kD) & (kH - 1);
    const float denom = ws[nAvg + b * kH + h];
    const float num   = ws[i];
    float val = (denom == 0.0f) ? 0.0f : num / denom;
    if (val != val) val = 1e-5f;   // NaN scrub
    out[i] = val;
  } else if (i < nAvg + kB * kH) {
    out[i] = ws[i];                // weights_sum passthrough
  }
}

extern "C" void kernel_launch(void* const* d_in, const int* in_sizes, int n_in,
                              void* d_out, int out_size, void* d_ws, size_t ws_size,
                              hipStream_t stream) {
  const float* feat = (const float*)d_in[0];   // [262144,128] f32
  const float* att  = (const float*)d_in[1];   // [262144,8]   f32
  const int*   seg  = (const int*)  d_in[2];   // [262144]     i32 (sorted)
  // d_in[3] = num_segments scalar (compile-time constant 64 here)

  float* ws  = (float*)d_ws;
  float* out = (float*)d_out;

  const int nWs = kB * kH * kD + kB * kH;      // 66048 f32 partials (~264 KB)
  zero_ws_kernel<<<(nWs + 255) / 256, 256, 0, stream>>>(ws, nWs);

  seg_wmma_kernel<<<kB * kSplits, kThreads, 0, stream>>>(feat, att, seg, ws);

  const int nOut = kB * kH * kD + kB * kH;     // 66048
  finalize_kernel<<<(nOut + 255) / 256, 256, 0, stream>>>(ws, out);
}